// CKConv_73856257622092
// MI455X (gfx1250) — compile-verified
//
#include <hip/hip_runtime.h>

// ---------------------------------------------------------------------------
// CKConv on MI455X (gfx1250, wave32)
//   B=4, CIN=32, COUT=64, T=2048, DK=16, K=2049
//   out[b,co,t] = bias[co] + sum_ci sum_{d=0..t} g[co,ci,d] * x[b,ci,t-d]
//   g[co,ci,d]  = kernel[co,ci,K-1-d] = w3[co*CIN+ci,:] . hh[:,d] + b3[..]
// Conv with V_WMMA_F32_16X16X4_F32 (fp32 in/accum): M=co, N=t, reduce (ci,lag).
// g stored lag-pair interleaved [d/2][ci][co][2] -> A fragment = one b64 load.
// Inner loop sweeps m = tile_base - d0; the two t-tiles of a wave share the
// same x (B) fragment, and one A fragment is carried across iterations so a
// single A load feeds two WMMAs.
// ---------------------------------------------------------------------------

#define B_    4
#define CIN   32
#define COUT  64
#define T_    2048
#define DK    16
#define KPOS  2049      // K = T + 1
#define DLEN  2048      // reachable lags d = 0..2047 (kernel[...,0] unused)

typedef float v2f __attribute__((ext_vector_type(2)));
typedef float v8f __attribute__((ext_vector_type(8)));

// --------------------------- kernel 1: SIREN -------------------------------
// hh[d*DK + i] = sin(om2*(w2 @ sin(om1*(w1*pos + b1)) + b2))[i],
// at pos = pos_rel[K-1-d]  (lag-reversed so the conv indexes g[d]).
__global__ void siren_kernel(const float* __restrict__ pos_rel,
                             const float* __restrict__ w1,
                             const float* __restrict__ b1,
                             const float* __restrict__ om1,
                             const float* __restrict__ w2,
                             const float* __restrict__ b2,
                             const float* __restrict__ om2,
                             float* __restrict__ hh) {
  int d = blockIdx.x * blockDim.x + threadIdx.x;
  if (d >= DLEN) return;
  float pos = pos_rel[KPOS - 1 - d];
  float o1 = om1[0], o2 = om2[0];
  float h1[DK];
#pragma unroll
  for (int j = 0; j < DK; ++j)
    h1[j] = __sinf(o1 * (w1[j] * pos + b1[j]));
#pragma unroll
  for (int i = 0; i < DK; ++i) {
    float a = b2[i];
#pragma unroll
    for (int j = 0; j < DK; ++j) a = fmaf(w2[i * DK + j], h1[j], a);
    hh[d * DK + i] = __sinf(o2 * a);
  }
}

// ------------------------ kernel 2: kernel gen -----------------------------
// Lag-pair-interleaved layout: g[(((d>>1)*CIN + ci)*COUT + co)*2 + (d&1)].
// Thread tid writes exactly flat element tid (coalesced stores):
//   parity=tid&1, co=(tid>>1)&63, ci=(tid>>7)&31, dp=tid>>12, d=2*dp+parity.
__global__ void kergen_kernel(const float* __restrict__ w3,
                              const float* __restrict__ b3,
                              const float* __restrict__ hh,
                              float* __restrict__ g) {
  int idx = blockIdx.x * blockDim.x + threadIdx.x;
  if (idx >= DLEN * CIN * COUT) return;
  int parity = idx & 1;
  int co = (idx >> 1) & (COUT - 1);
  int ci = (idx >> 7) & (CIN - 1);
  int dp = idx >> 12;
  int d  = 2 * dp + parity;
  int r  = co * CIN + ci;                 // w3 row: kernel.reshape(COUT,CIN,K)
  const float* wr = w3 + r * DK;
  const float* hr = hh + d * DK;
  float a = b3[r];
#pragma unroll
  for (int j = 0; j < DK; ++j) a = fmaf(wr[j], hr[j], a);
  g[idx] = a;
}

// ------------------------- kernel 3: WMMA conv -----------------------------
// Block = (batch b, 32-wide t-super-tile), 4 waves; wave w owns co-tile
// co0 = 16*w and TWO t-tiles (t0, t0+16).
__global__ __launch_bounds__(128) void conv_wmma_kernel(
    const float* __restrict__ x,     // (B, CIN, T)
    const float* __restrict__ g,     // interleaved (DLEN/2, CIN, COUT, 2)
    const float* __restrict__ bias,  // (COUT)
    float* __restrict__ out) {       // (B, COUT, T)
  const int wave = threadIdx.x >> 5;
  const int lane = threadIdx.x & 31;       // wave32
  const int n    = lane & 15;
  const int half = lane >> 4;              // 0: lanes 0-15, 1: lanes 16-31
  const int bb   = blockIdx.x / (T_ / 32);
  const int t0   = (blockIdx.x % (T_ / 32)) * 32;  // tiles at t0, t0+16
  const int co0  = wave * 16;

  v8f acc0 = {0.f, 0.f, 0.f, 0.f, 0.f, 0.f, 0.f, 0.f};
  v8f acc1 = {0.f, 0.f, 0.f, 0.f, 0.f, 0.f, 0.f, 0.f};

  const float* xb = x + bb * (CIN * T_);
  const v2f*   gv = (const v2f*)g;         // v2f units: [(d>>1)][ci][co]
  const int lanegA = co0 + n;              // per-lane co component
  const int MSTEP  = 8 * CIN * COUT;       // d0 -= 16  =>  v2f index -= 8*CIN*COUT

  for (int ci = 0; ci < CIN; ++ci) {
    const float* xc = xb + ci * T_;
#pragma unroll
    for (int c = 0; c < 4; ++c) {
      const int di = c * 4 + 2 * half;     // this lane: v0 <-> kk=2*half, v1 <-> +1
      // ---- peel m = 0 (only iteration where causal clamp can fire) ----
      int s0 = n - di;
      v2f bfrag;
      bfrag.x = (s0 >= 0) ? xc[s0] : 0.0f;
      bfrag.y = (s0 >= 1) ? xc[s0 - 1] : 0.0f;
      // A fragment (v2f units): d-pair base (d0 + 4c + 2*half) >> 1
      int idxA = (((t0 + 16) >> 1) + 2 * c + half) * (CIN * COUT)
                 + ci * COUT + lanegA;     // d0 = t0 + 16  (acc1)
      v2f a1 = gv[idxA];
      acc1 = __builtin_amdgcn_wmma_f32_16x16x4_f32(
          false, a1, false, bfrag, (short)0, acc1, false, false);
      idxA -= MSTEP;                       // d0 = t0       (acc0)
      v2f aPrev = gv[idxA];
      acc0 = __builtin_amdgcn_wmma_f32_16x16x4_f32(
          false, aPrev, false, bfrag, (short)0, acc0, false, false);
      // ---- steady state: m = 16 .. t0+16, no clamps ----
      for (int m = 16; m <= t0 + 16; m += 16) {
        s0 = m + n - di;                   // <= t0+31 <= 2047 < T, >= 1
        bfrag.x = xc[s0];
        bfrag.y = xc[s0 - 1];
        // acc1 at d0 = t0+16-m == acc0's block from previous step: reuse aPrev
        acc1 = __builtin_amdgcn_wmma_f32_16x16x4_f32(
            false, aPrev, false, bfrag, (short)0, acc1, false, false);
        if (m <= t0) {                     // wave-uniform: EXEC stays all-1s
          idxA -= MSTEP;                   // d0 = t0 - m   (acc0)
          v2f aNew = gv[idxA];
          acc0 = __builtin_amdgcn_wmma_f32_16x16x4_f32(
              false, aNew, false, bfrag, (short)0, acc0, false, false);
          aPrev = aNew;
        }
      }
    }
  }

  // D layout: VGPR r holds D[M = r + 8*half][N = n];  M -> co, N -> t.
  float* ob = out + (bb * COUT + co0) * T_;
#pragma unroll
  for (int r = 0; r < 8; ++r) {
    int co = r + 8 * half;
    float bv = bias[co0 + co];
    ob[co * T_ + t0 + n]      = acc0[r] + bv;
    ob[co * T_ + t0 + 16 + n] = acc1[r] + bv;
  }
}

// ------------------------------- launch ------------------------------------
extern "C" void kernel_launch(void* const* d_in, const int* in_sizes, int n_in,
                              void* d_out, int out_size, void* d_ws, size_t ws_size,
                              hipStream_t stream) {
  const float* x       = (const float*)d_in[0];   // (B, CIN, T)
  const float* pos_rel = (const float*)d_in[1];   // (K,)
  const float* w1      = (const float*)d_in[2];   // (DK, 1)
  const float* b1      = (const float*)d_in[3];   // (DK,)
  const float* om1     = (const float*)d_in[4];   // scalar
  const float* w2      = (const float*)d_in[5];   // (DK, DK)
  const float* b2      = (const float*)d_in[6];   // (DK,)
  const float* om2     = (const float*)d_in[7];   // scalar
  const float* w3      = (const float*)d_in[8];   // (COUT*CIN, DK)
  const float* b3      = (const float*)d_in[9];   // (COUT*CIN,)
  const float* bias    = (const float*)d_in[10];  // (COUT,)
  float* out = (float*)d_out;

  // Workspace: hh (2048*16 f32 = 128KB) then g (2048*32*64 f32 = 16MB).
  float* hh = (float*)d_ws;
  float* g  = hh + DLEN * DK;

  siren_kernel<<<(DLEN + 255) / 256, 256, 0, stream>>>(
      pos_rel, w1, b1, om1, w2, b2, om2, hh);

  kergen_kernel<<<(DLEN * CIN * COUT) / 256, 256, 0, stream>>>(w3, b3, hh, g);

  conv_wmma_kernel<<<B_ * (T_ / 32), 128, 0, stream>>>(x, g, bias, out);
}